// GraphPooling_57664230916668
// MI455X (gfx1250) — compile-verified
//
#include <hip/hip_runtime.h>

// ---------------------------------------------------------------------------
// GraphPooling: h_new[n] = mean over in-neighbors of h, keep h where deg==0.
//   B=8, N=8192, F=64  -> num_nodes = 65536
//   E = 1,048,576 edges (int32 index arrays per harness convention)
//
// Plan (MI455X):
//   1) memsetAsync workspace:   msum[65536*64] f32, deg[65536] f32   (16.25 MB,
//      L2-resident on the 192MB L2 -> all edge traffic stays on-chip)
//   2) scatter kernel: one wave32 per edge; lane l adds h[src*64 + 2l .. 2l+1]
//      into msum[dst*64 + ...] with native global_atomic_add_f32; lane 0 bumps deg.
//   3) finalize kernel: per wave, a 16-node x 16-feature tile:
//      mean_tile = diag(1/deg) x msum_tile  via V_WMMA_F32_16X16X4_F32
//      (4 accumulating K=4 chunks), then select h where deg==0.
// ---------------------------------------------------------------------------

typedef __attribute__((ext_vector_type(2))) float v2f;
typedef __attribute__((ext_vector_type(8))) float v8f;

#define NUM_NODES (8 * 8192)
#define FDIM 64

__global__ void gp_scatter_kernel(const float* __restrict__ h,
                                  const int* __restrict__ src,
                                  const int* __restrict__ dst,
                                  float* __restrict__ msum,
                                  float* __restrict__ deg,
                                  int num_edges) {
    const int lane   = threadIdx.x & 31;
    const int wave   = (blockIdx.x * blockDim.x + threadIdx.x) >> 5;
    const int nwaves = (gridDim.x * blockDim.x) >> 5;

    for (int e = wave; e < num_edges; e += nwaves) {
        const int s = src[e];
        const int d = dst[e];
        // 256B coalesced gather: lane l takes features 2l, 2l+1
        const float2 hv = *(const float2*)(h + (size_t)s * FDIM + lane * 2);
        float* mrow = msum + (size_t)d * FDIM + lane * 2;
        unsafeAtomicAdd(mrow + 0, hv.x);   // global_atomic_add_f32 (L2-resident)
        unsafeAtomicAdd(mrow + 1, hv.y);
        if (lane == 0) unsafeAtomicAdd(deg + d, 1.0f);
    }
}

// One wave per 16-node tile; wave loops over 4 feature chunks of 16.
// WMMA layouts (CDNA5 ISA 7.12.2, fp32, wave32):
//   A 16x4 : lanes 0-15 -> M=lane, VGPR0=K0, VGPR1=K1 ; lanes 16-31 -> K2,K3
//   B 4x16 : lanes 0-15 -> N=lane, VGPR0=K0, VGPR1=K1 ; lanes 16-31 -> K2,K3
//   C/D    : lanes 0-15 -> (M=v, N=lane), lanes 16-31 -> (M=v+8, N=lane-16)
__global__ void gp_finalize_kernel(const float* __restrict__ h,
                                   const float* __restrict__ msum,
                                   const float* __restrict__ deg,
                                   float* __restrict__ out,
                                   int num_tiles) {
    const int lane = threadIdx.x & 31;
    const int wave = (blockIdx.x * blockDim.x + threadIdx.x) >> 5;
    if (wave >= num_tiles) return;          // uniform per wave: EXEC stays all-1s

    const int node_base = wave * 16;
    const int half      = lane >> 4;        // 0: lanes 0-15, 1: lanes 16-31
    const int m5        = lane & 15;

    // reciprocal degree for this lane's diagonal row (m = m5)
    const float dm = deg[node_base + m5];
    const float r  = (dm > 0.0f) ? (1.0f / dm) : 1.0f;  // deg==0 rows: msum==0 anyway

    // degrees of the 8 output rows this lane stores (rows half*8 .. half*8+7)
    const int rowbase = half * 8;
    float degrow[8];
#pragma unroll
    for (int v = 0; v < 8; ++v) degrow[v] = deg[node_base + rowbase + v];

    for (int fc = 0; fc < 4; ++fc) {
        const int feat = fc * 16 + m5;      // this lane's column
        v8f c = {};
#pragma unroll
        for (int j = 0; j < 4; ++j) {       // K-chunks of the 16x16 diagonal
            v2f a;
            a.x = (m5 == 4 * j + 2 * half + 0) ? r : 0.0f;
            a.y = (m5 == 4 * j + 2 * half + 1) ? r : 0.0f;
            const int krow = node_base + 4 * j + 2 * half;
            v2f b;
            b.x = msum[(size_t)(krow + 0) * FDIM + feat];
            b.y = msum[(size_t)(krow + 1) * FDIM + feat];
            // D = A x B + C   (fp32 WMMA, round-to-nearest-even)
            c = __builtin_amdgcn_wmma_f32_16x16x4_f32(
                    /*neg_a=*/false, a, /*neg_b=*/false, b,
                    /*c_mod=*/(short)0, c, /*reuse_a=*/false, /*reuse_b=*/false);
        }
#pragma unroll
        for (int v = 0; v < 8; ++v) {
            const size_t idx = (size_t)(node_base + rowbase + v) * FDIM + feat;
            const float hv = h[idx];
            out[idx] = (degrow[v] > 0.0f) ? c[v] : hv;
        }
    }
}

extern "C" void kernel_launch(void* const* d_in, const int* in_sizes, int n_in,
                              void* d_out, int out_size, void* d_ws, size_t ws_size,
                              hipStream_t stream) {
    const float* h   = (const float*)d_in[0];
    const int*   src = (const int*)d_in[1];
    const int*   dst = (const int*)d_in[2];
    float*       out = (float*)d_out;
    const int num_edges = in_sizes[1];

    float* msum = (float*)d_ws;
    float* deg  = msum + (size_t)NUM_NODES * FDIM;

    // zero accumulators every call (harness poisons d_ws once, never re-poisons)
    hipMemsetAsync(d_ws, 0, ((size_t)NUM_NODES * FDIM + NUM_NODES) * sizeof(float),
                   stream);

    // 16384 waves -> 64 edges/wave; 8 waves per 256-thread block
    gp_scatter_kernel<<<2048, 256, 0, stream>>>(h, src, dst, msum, deg, num_edges);

    // 65536/16 = 4096 node tiles, one wave each -> 512 blocks of 8 waves
    const int num_tiles = NUM_NODES / 16;
    gp_finalize_kernel<<<num_tiles / 8, 256, 0, stream>>>(h, msum, deg, out, num_tiles);
}